// GATLayer_76510547411436
// MI455X (gfx1250) — compile-verified
//
#include <hip/hip_runtime.h>
#include <math.h>

#define NN 50000           // 3125 * 16 exactly -> no GEMM row guards needed
#define NE 800000          // 3125 * 256 exactly -> no edge guards needed
#define DD 128
#define KDIM 256           // 2*D
#define XS_STRIDE (KDIM + 4)

typedef __attribute__((ext_vector_type(2))) float v2f;
typedef __attribute__((ext_vector_type(8))) float v8f;

__device__ __forceinline__ float wave_reduce_add(float v) {
  v += __shfl_xor(v, 16, 32);
  v += __shfl_xor(v, 8, 32);
  v += __shfl_xor(v, 4, 32);
  v += __shfl_xor(v, 2, 32);
  v += __shfl_xor(v, 1, 32);
  return v;
}

// ---------------------------------------------------------------------------
// Kernel 1: per-node attention dots s[i]=h[i].aw_src, t[i]=h[i].aw_dst,
//           init emax bits (0 == +0.0f, valid identity since e>=0 after relu),
//           init denom, zero z. One wave32 per node; grid*8 == NN exactly.
// ---------------------------------------------------------------------------
__global__ void node_prep(const float* __restrict__ h,
                          const float* __restrict__ attn_w,
                          float* __restrict__ s, float* __restrict__ t,
                          unsigned int* __restrict__ emax_bits,
                          float* __restrict__ denom,
                          float* __restrict__ z) {
  int node = (blockIdx.x * blockDim.x + threadIdx.x) >> 5;
  int lane = threadIdx.x & 31;
  float4 hx = ((const float4*)(h + (size_t)node * DD))[lane];
  float4 a  = ((const float4*)(attn_w))[lane];        // aw_src chunk
  float4 b  = ((const float4*)(attn_w + DD))[lane];   // aw_dst chunk
  float ss = hx.x * a.x + hx.y * a.y + hx.z * a.z + hx.w * a.w;
  float tt = hx.x * b.x + hx.y * b.y + hx.z * b.z + hx.w * b.w;
  ss = wave_reduce_add(ss);
  tt = wave_reduce_add(tt);
  if (lane == 0) {
    s[node] = ss;
    t[node] = tt;
    emax_bits[node] = 0u;   // +0.0f
    denom[node] = 0.0f;
  }
  float4 z4 = {0.f, 0.f, 0.f, 0.f};
  ((float4*)(z + (size_t)node * DD))[lane] = z4;
}

// ---------------------------------------------------------------------------
// Kernel 2: segment max of e over dst. e>=0 so uint-bit atomicMax is exact.
// ---------------------------------------------------------------------------
__global__ void edge_max(const float* __restrict__ s, const float* __restrict__ t,
                         const int* __restrict__ src, const int* __restrict__ dst,
                         const float* __restrict__ attn_b,
                         unsigned int* __restrict__ emax_bits) {
  int k = blockIdx.x * blockDim.x + threadIdx.x;
  int d = dst[k];
  float e = fmaxf(s[src[k]] + t[d] + attn_b[0], 0.0f);
  atomicMax(&emax_bits[d], __float_as_uint(e));
}

// ---------------------------------------------------------------------------
// Kernel 3: denom[dst] += exp(e - emax[dst]) (recompute e: 2 gathers, cheap)
// ---------------------------------------------------------------------------
__global__ void edge_denom(const float* __restrict__ s, const float* __restrict__ t,
                           const int* __restrict__ src, const int* __restrict__ dst,
                           const float* __restrict__ attn_b,
                           const unsigned int* __restrict__ emax_bits,
                           float* __restrict__ denom) {
  int k = blockIdx.x * blockDim.x + threadIdx.x;
  int d = dst[k];
  float e = fmaxf(s[src[k]] + t[d] + attn_b[0], 0.0f);
  float ex = expf(e - __uint_as_float(emax_bits[d]));
  atomicAdd(&denom[d], ex);
}

// ---------------------------------------------------------------------------
// Kernel 4: z[dst] += alpha * efeats[edge]. One wave32 per edge, float4/lane.
// z (25.6 MB) is L2-resident (192 MB L2) -> atomics resolve on-chip; the
// 409.6 MB efeats stream at 23.3 TB/s is the true floor of this layer.
// ---------------------------------------------------------------------------
__global__ void edge_scatter(const float* __restrict__ s, const float* __restrict__ t,
                             const int* __restrict__ src, const int* __restrict__ dst,
                             const float* __restrict__ attn_b,
                             const unsigned int* __restrict__ emax_bits,
                             const float* __restrict__ denom,
                             const float* __restrict__ ef,
                             float* __restrict__ z) {
  int edge = (blockIdx.x * blockDim.x + threadIdx.x) >> 5;
  int lane = threadIdx.x & 31;
  int d = dst[edge];
  float e = fmaxf(s[src[edge]] + t[d] + attn_b[0], 0.0f);
  float alpha = expf(e - __uint_as_float(emax_bits[d])) / denom[d];
  float4 ev = ((const float4*)(ef + (size_t)edge * DD))[lane];
  float* zp = z + (size_t)d * DD + lane * 4;
  atomicAdd(zp + 0, alpha * ev.x);
  atomicAdd(zp + 1, alpha * ev.y);
  atomicAdd(zp + 2, alpha * ev.z);
  atomicAdd(zp + 3, alpha * ev.w);
}

// ---------------------------------------------------------------------------
// Kernel 5: out = relu([h|z] @ W^T + b) via V_WMMA_F32_16X16X4_F32 (fp32 exact).
// Block = 256 threads = 8 waves; block owns a 16-row node tile (3125 blocks,
// exact); wave w owns output columns [16w,16w+16). X tile (16x256) staged in
// LDS (stride 260 floats -> conflict-free: lane banks 4r+col). K-loop: 64
// WMMAs of K=4. Fragment layouts per CDNA5 ISA 7.12.2:
//   A (16x4 f32, 2 VGPR): v0 = K0 (lanes 0-15) / K2 (lanes 16-31), v1 = K1/K3
//   B (4x16 f32, 2 VGPR): v0 = rows K0/K2 striped over lanes, v1 = K1/K3
//   C (16x16 f32, 8 VGPR): vr = row r (lanes 0-15) / row r+8 (lanes 16-31)
// ---------------------------------------------------------------------------
__global__ void out_gemm(const float* __restrict__ h, const float* __restrict__ z,
                         const float* __restrict__ W, const float* __restrict__ bias,
                         float* __restrict__ out) {
  __shared__ float Xs[16][XS_STRIDE];
  const int base = blockIdx.x * 16;
  const int tid  = threadIdx.x;

  // Stage X = [h | z] tile: 16 rows x 256 cols; each thread loads 16 floats.
  {
    int row  = tid >> 4;          // 0..15
    int col0 = (tid & 15) * 16;   // 0,16,...,240 (chunk never straddles 128)
    int node = base + row;
    const float* p = (col0 < DD) ? (h + (size_t)node * DD + col0)
                                 : (z + (size_t)node * DD + (col0 - DD));
    float4 v0 = ((const float4*)p)[0];
    float4 v1 = ((const float4*)p)[1];
    float4 v2 = ((const float4*)p)[2];
    float4 v3 = ((const float4*)p)[3];
    float4* q = (float4*)&Xs[row][col0];
    q[0] = v0; q[1] = v1; q[2] = v2; q[3] = v3;
  }
  __syncthreads();

  const int wv   = tid >> 5;          // wave id 0..7 -> column tile
  const int lane = tid & 31;
  const int m    = lane & 15;         // A row
  const int jn   = lane & 15;         // B/C column within tile
  const int koff = (lane >> 4) << 1;  // half-wave K split: 0 or 2
  const float* Wrow = W + (size_t)(wv * 16 + jn) * KDIM;

  v8f c = {0.f, 0.f, 0.f, 0.f, 0.f, 0.f, 0.f, 0.f};
#pragma unroll 8
  for (int k0 = 0; k0 < KDIM; k0 += 4) {
    v2f a, b;
    a.x = Xs[m][k0 + koff];
    a.y = Xs[m][k0 + koff + 1];
    b.x = Wrow[k0 + koff];
    b.y = Wrow[k0 + koff + 1];
    c = __builtin_amdgcn_wmma_f32_16x16x4_f32(
        /*neg_a=*/false, a, /*neg_b=*/false, b,
        /*c_mod=*/(short)0, c, /*reuse_a=*/false, /*reuse_b=*/false);
  }

  // Epilogue: bias + relu. NN % 16 == 0 -> no row guards; strides fold into
  // immediate store offsets off one base address.
  const float bj = bias[wv * 16 + jn];
  const int mb = (lane >> 4) << 3;    // 0 or 8: upper half-wave holds rows +8
  float* op = out + (size_t)(base + mb) * DD + wv * 16 + jn;
#pragma unroll
  for (int r = 0; r < 8; ++r) {
    op[(size_t)r * DD] = fmaxf(c[r] + bj, 0.0f);
  }
}

// ---------------------------------------------------------------------------
extern "C" void kernel_launch(void* const* d_in, const int* in_sizes, int n_in,
                              void* d_out, int out_size, void* d_ws, size_t ws_size,
                              hipStream_t stream) {
  const float* nfeats = (const float*)d_in[0];   // (NN,1,128) f32
  const float* efeats = (const float*)d_in[1];   // (NE,1,128) f32
  const float* Wap    = (const float*)d_in[2];   // (128,256)  f32
  const float* Wab    = (const float*)d_in[3];   // (128,)     f32
  const float* attn_w = (const float*)d_in[4];   // (1,256)    f32
  const float* attn_b = (const float*)d_in[5];   // (1,)       f32
  const int*   src    = (const int*)d_in[6];     // (NE,) i32
  const int*   dst    = (const int*)d_in[7];     // (NE,) i32
  float* out = (float*)d_out;                    // (NN,1,128) f32

  // Workspace carve-out (256B aligned): s, t, emax_bits, denom, z (~26.5 MB)
  char* ws = (char*)d_ws;
  size_t off = 0;
  auto carve = [&](size_t bytes) {
    void* p = ws + off;
    off += (bytes + 255) & ~(size_t)255;
    return p;
  };
  float*        s_buf  = (float*)carve((size_t)NN * sizeof(float));
  float*        t_buf  = (float*)carve((size_t)NN * sizeof(float));
  unsigned int* emax_b = (unsigned int*)carve((size_t)NN * sizeof(unsigned int));
  float*        denom  = (float*)carve((size_t)NN * sizeof(float));
  float*        z_buf  = (float*)carve((size_t)NN * DD * sizeof(float));
  (void)ws_size; (void)in_sizes; (void)n_in; (void)out_size;

  const int B = 256;
  // 1) per-node prep: wave per node (8 nodes/block, 6250*8 == NN)
  node_prep<<<NN / 8, B, 0, stream>>>(nfeats, attn_w, s_buf, t_buf,
                                      emax_b, denom, z_buf);
  // 2) segment max (3125*256 == NE)
  edge_max<<<NE / B, B, 0, stream>>>(s_buf, t_buf, src, dst, attn_b, emax_b);
  // 3) segment sum of exp
  edge_denom<<<NE / B, B, 0, stream>>>(s_buf, t_buf, src, dst, attn_b,
                                       emax_b, denom);
  // 4) weighted scatter of edge features: wave per edge (100000*8 == NE)
  edge_scatter<<<NE / 8, B, 0, stream>>>(s_buf, t_buf, src, dst, attn_b,
                                         emax_b, denom, efeats, z_buf);
  // 5) output projection with f32 WMMA: 16-row tile per block (3125*16 == NN)
  out_gemm<<<NN / 16, B, 0, stream>>>(nfeats, z_buf, Wap, Wab, out);
}